// GNNBaseline_6262062317940
// MI455X (gfx1250) — compile-verified
//
#include <hip/hip_runtime.h>

typedef __attribute__((ext_vector_type(2))) float v2f;
typedef __attribute__((ext_vector_type(8))) float v8f;

#define H_DIM 128
#define KC    64          // K-chunk of W staged in LDS
#define PITCH 144         // float2 per K-pair row: 2*PITCH % 64 == 32 -> half-waves on disjoint banks

__device__ __forceinline__ void atomAddF(float* p, float v) {
    // agent-scope relaxed f32 add -> global_atomic_add_f32 (no return)
    __hip_atomic_fetch_add(p, v, __ATOMIC_RELAXED, __HIP_MEMORY_SCOPE_AGENT);
}

// ---------------- degree / dinv -------------------------------------------
__global__ void gcn_deg_init(float* __restrict__ deg, int N) {
    int i = blockIdx.x * blockDim.x + threadIdx.x;
    if (i < N) deg[i] = 1.0f;                 // self-loop contribution
}
__global__ void gcn_deg_accum(const int* __restrict__ dst, float* __restrict__ deg, int E) {
    int e = blockIdx.x * blockDim.x + threadIdx.x;
    if (e < E) atomAddF(&deg[dst[e]], 1.0f);
}
__global__ void gcn_deg_rsqrt(float* __restrict__ deg, int N) {
    int i = blockIdx.x * blockDim.x + threadIdx.x;
    if (i < N) deg[i] = rsqrtf(deg[i]);
}

// ---------------- bias init (scatter accumulator = b), float4 stores ------
__global__ void gcn_bias_init(float4* __restrict__ h, const float4* __restrict__ b4,
                              long long total4) {
    long long i = (long long)blockIdx.x * blockDim.x + threadIdx.x;
    if (i < total4) h[i] = b4[i & (H_DIM / 4 - 1)];
}

// ---------------- fp32 WMMA GEMM:  Out[N,128] = act(A[N,K]) @ W[K,128] ----
// wave = one 16-row strip x all 8 column tiles; W chunk lives in LDS as
// K-pair float2 so each B fragment is a single ds_load_b64.
template <int K, bool RELU>
__global__ __launch_bounds__(256)
void gcn_gemm_wmma(const float* __restrict__ A, const float* __restrict__ W,
                   float* __restrict__ Out, int N) {
    __shared__ v2f Wlds[(KC / 2) * PITCH];    // 36 KB
    const int lane  = threadIdx.x & 31;
    const int wave  = threadIdx.x >> 5;
    const int l     = lane & 15;              // row (A) / col (B,D) within tile
    const int khalf = lane >> 4;              // 0: K=k0,k0+1   1: K=k0+2,k0+3
    const long long row0 = ((long long)blockIdx.x * 8 + wave) * 16;
    const bool active = (row0 + 16) <= (long long)N;   // wave-uniform; N%16==0

    v8f acc[8];
#pragma unroll
    for (int c = 0; c < 8; ++c)
#pragma unroll
        for (int r = 0; r < 8; ++r) acc[c][r] = 0.0f;

    for (int kc = 0; kc < K; kc += KC) {
        // stage W[kc..kc+KC) x 128 as {W[k][c], W[k+1][c]} pairs
        for (int i = threadIdx.x; i < (KC / 2) * H_DIM; i += 256) {
            const int p  = i >> 7;            // K-pair index within chunk
            const int cc = i & (H_DIM - 1);   // output column
            v2f w;
            w.x = W[(long long)(kc + 2 * p)     * H_DIM + cc];
            w.y = W[(long long)(kc + 2 * p + 1) * H_DIM + cc];
            Wlds[p * PITCH + cc] = w;
        }
        __syncthreads();

        if (active) {
            const float* __restrict__ Arow = A + row0 * K;
#pragma unroll 4
            for (int k = 0; k < KC; k += 4) {
                // A fragment: one aligned 8-byte load per lane
                v2f av = *(const v2f*)(Arow + (long long)l * K + (kc + k + khalf * 2));
                if (RELU) { av.x = fmaxf(av.x, 0.0f); av.y = fmaxf(av.y, 0.0f); }
                const int p = (k >> 1) + khalf;       // K-pair row in LDS
#pragma unroll
                for (int c = 0; c < 8; ++c) {
                    const v2f bv = Wlds[p * PITCH + c * 16 + l];   // ds_load_b64
                    acc[c] = __builtin_amdgcn_wmma_f32_16x16x4_f32(
                        false, av, false, bv, (short)0, acc[c], false, false);
                }
            }
        }
        __syncthreads();
    }

    if (active) {
        // D layout: VGPR r -> row = r (lanes 0-15) / r+8 (lanes 16-31), col = l
#pragma unroll
        for (int c = 0; c < 8; ++c)
#pragma unroll
            for (int r = 0; r < 8; ++r)
                Out[(row0 + r + khalf * 8) * H_DIM + c * 16 + l] = acc[c][r];
    }
}

// ---------------- gather * norm -> atomic scatter-add ---------------------
__global__ __launch_bounds__(256)
void gcn_propagate(const float* __restrict__ hlin, float* __restrict__ hout,
                   const int* __restrict__ src, const int* __restrict__ dst,
                   const float* __restrict__ dinv, int E, int N) {
    long long e = (long long)blockIdx.x * 8 + (threadIdx.x >> 5);
    const long long TE = (long long)E + N;
    if (e >= TE) return;
    const int lane = threadIdx.x & 31;
    int s, d;
    if (e < E) { s = src[e]; d = dst[e]; } else { s = d = (int)(e - E); }
    const float norm = dinv[s] * dinv[d];
    const float4 m = ((const float4*)(hlin + (long long)s * H_DIM))[lane];
    float* o = hout + (long long)d * H_DIM + lane * 4;
    atomAddF(o + 0, m.x * norm);
    atomAddF(o + 1, m.y * norm);
    atomAddF(o + 2, m.z * norm);
    atomAddF(o + 3, m.w * norm);
}

// ---------------- global mean pool + head ---------------------------------
__global__ void gcn_zero(float* __restrict__ p, int n) {
    int i = blockIdx.x * blockDim.x + threadIdx.x;
    if (i < n) p[i] = 0.0f;
}
__global__ __launch_bounds__(256)
void gcn_pool_accum(const float* __restrict__ h, const int* __restrict__ batch,
                    float* __restrict__ pooled, float* __restrict__ cnt, int N) {
    long long n = (long long)blockIdx.x * 8 + (threadIdx.x >> 5);
    if (n >= N) return;
    const int lane = threadIdx.x & 31;
    const int g = batch[n];
    const float4 m = ((const float4*)(h + n * H_DIM))[lane];
    float* o = pooled + (long long)g * H_DIM + lane * 4;
    atomAddF(o + 0, m.x); atomAddF(o + 1, m.y);
    atomAddF(o + 2, m.z); atomAddF(o + 3, m.w);
    if (lane == 0) atomAddF(&cnt[g], 1.0f);
}
__global__ void gcn_head(const float* __restrict__ pooled, const float* __restrict__ cnt,
                         const float* __restrict__ Wc, const float* __restrict__ bc,
                         float* __restrict__ out) {
    int t = threadIdx.x;                 // 128 threads: g = t>>1, c = t&1
    if (t >= 128) return;
    const int g = t >> 1, c = t & 1;
    const float inv = 1.0f / fmaxf(cnt[g], 1.0f);
    float acc = bc[c];
    for (int f = 0; f < H_DIM; ++f)
        acc = fmaf(pooled[g * H_DIM + f] * inv, Wc[f * 2 + c], acc);
    out[g * 2 + c] = acc;
}

// ---------------- driver ---------------------------------------------------
extern "C" void kernel_launch(void* const* d_in, const int* in_sizes, int n_in,
                              void* d_out, int out_size, void* d_ws, size_t ws_size,
                              hipStream_t stream) {
    const float* x    = (const float*)d_in[0];
    const int*   eidx = (const int*)d_in[1];
    const int*   batch= (const int*)d_in[2];
    const float* W1   = (const float*)d_in[3];
    const float* b1   = (const float*)d_in[4];
    const float* W2   = (const float*)d_in[5];
    const float* b2   = (const float*)d_in[6];
    const float* W3   = (const float*)d_in[7];
    const float* b3   = (const float*)d_in[8];
    const float* Wc   = (const float*)d_in[9];
    const float* bc   = (const float*)d_in[10];

    const int N = in_sizes[0] / 256;
    const int E = in_sizes[1] / 2;
    const int* srcI = eidx;
    const int* dstI = eidx + E;

    float* wsf = (float*)d_ws;
    size_t o = 0;
    float* dinv   = wsf + o; o += (size_t)((N + 127) / 128) * 128;
    float* bufA   = wsf + o; o += (size_t)N * H_DIM;
    float* bufB   = wsf + o; o += (size_t)N * H_DIM;
    float* pooled = wsf + o; o += 64 * H_DIM;
    float* cnt    = wsf + o; o += 64;
    (void)ws_size; (void)n_in; (void)out_size;

    const long long NH4 = (long long)N * (H_DIM / 4);      // float4 elements
    const int blkN   = (N + 255) / 256;
    const int blkE   = (E + 255) / 256;
    const int blkNH4 = (int)((NH4 + 255) / 256);
    const int blkGEM = ((N + 15) / 16 + 7) / 8;
    const int blkPRP = (int)(((long long)E + N + 7) / 8);
    const int blkPOL = (N + 7) / 8;

    // symmetric normalization: deg = 1 + indegree ; dinv = rsqrt(deg)
    gcn_deg_init <<<blkN, 256, 0, stream>>>(dinv, N);
    gcn_deg_accum<<<blkE, 256, 0, stream>>>(dstI, dinv, E);
    gcn_deg_rsqrt<<<blkN, 256, 0, stream>>>(dinv, N);

    // layer 1: hlin = x @ W1 ; hout = b1 + scatter(norm * gather(hlin))
    gcn_gemm_wmma<256, false><<<blkGEM, 256, 0, stream>>>(x, W1, bufA, N);
    gcn_bias_init<<<blkNH4, 256, 0, stream>>>((float4*)bufB, (const float4*)b1, NH4);
    gcn_propagate<<<blkPRP, 256, 0, stream>>>(bufA, bufB, srcI, dstI, dinv, E, N);

    // layer 2 (ReLU fused into A-read)
    gcn_gemm_wmma<128, true><<<blkGEM, 256, 0, stream>>>(bufB, W2, bufA, N);
    gcn_bias_init<<<blkNH4, 256, 0, stream>>>((float4*)bufB, (const float4*)b2, NH4);
    gcn_propagate<<<blkPRP, 256, 0, stream>>>(bufA, bufB, srcI, dstI, dinv, E, N);

    // layer 3 (ReLU fused, no output ReLU)
    gcn_gemm_wmma<128, true><<<blkGEM, 256, 0, stream>>>(bufB, W3, bufA, N);
    gcn_bias_init<<<blkNH4, 256, 0, stream>>>((float4*)bufB, (const float4*)b3, NH4);
    gcn_propagate<<<blkPRP, 256, 0, stream>>>(bufA, bufB, srcI, dstI, dinv, E, N);

    // mean pool + classifier head
    gcn_zero      <<<(64 * H_DIM + 64 + 255) / 256, 256, 0, stream>>>(pooled, 64 * H_DIM + 64);
    gcn_pool_accum<<<blkPOL, 256, 0, stream>>>(bufB, batch, pooled, cnt, N);
    gcn_head      <<<1, 128, 0, stream>>>(pooled, cnt, Wc, bc, (float*)d_out);
}